// Model_80968723464861
// MI455X (gfx1250) — compile-verified
//
#include <hip/hip_runtime.h>

typedef __attribute__((ext_vector_type(2))) float v2f;
typedef __attribute__((ext_vector_type(4))) float v4f;
typedef __attribute__((ext_vector_type(8))) float v8f;

namespace {
constexpr int   kNet   = 512;
constexpr int   kDimIn = 513;
constexpr int   kB     = 256;
constexpr int   kT     = 300;
constexpr int   kDimOut = 2;
constexpr int   kChunks = kNet / 4;                  // 128 K-chunks of 4
constexpr float kA     = 0.1f;                       // DT / TAU
constexpr float kTwoPiOverNet = 6.28318530717958647692f / 512.0f;
}

// ---------------------------------------------------------------------------
// Phase 1: input projection. W_x is circulant -> rank-2:
//   Cin[row] = sum_{d<512} input[row][d] * cos(2*pi*d/512)
//   Sin[row] = sum_{d<512} input[row][d] * sin(2*pi*d/512)
// Done as a [rows x 512] x [512 x 2] f32 GEMM with V_WMMA_F32_16X16X4_F32.
// Each wave owns a 16-row tile; per WMMA each lane loads exactly its two
// A-operand dwords (lanes 0-15: K=k,k+1 ; lanes 16-31: K=k+2,k+3), so WMMA
// issue == streaming rate. The B operand (cos in col 0, sin in col 1, zeros
// elsewhere) is precomputed into LDS in the per-lane WMMA register layout,
// so the inner loop is branch-free: global_b64 + ds_b64 + wmma.
// ---------------------------------------------------------------------------
__global__ __launch_bounds__(256) void inproj_wmma_kernel(
    const float* __restrict__ input, float* __restrict__ Cin,
    float* __restrict__ Sin) {
  // bTab[chunk][lane] = the v2f this lane feeds as B for this K-chunk.
  __shared__ v2f bTab[kChunks * 32];
  const int tid = threadIdx.x;

  // Fill B table once (uniform, no divergence in the hot loop).
  for (int idx = tid; idx < kChunks * 32; idx += 256) {
    const int c = idx >> 5;          // K-chunk
    const int l = idx & 31;          // lane this entry belongs to
    const int col = l & 15;          // B column
    const int k = 4 * c + ((l >> 4) << 1);  // K row pair base for this lane
    v2f val;
    if (col == 0) {
      val.x = cosf((float)k * kTwoPiOverNet);
      val.y = cosf((float)(k + 1) * kTwoPiOverNet);
    } else if (col == 1) {
      val.x = sinf((float)k * kTwoPiOverNet);
      val.y = sinf((float)(k + 1) * kTwoPiOverNet);
    } else {
      val.x = 0.0f;
      val.y = 0.0f;
    }
    bTab[idx] = val;
  }
  __syncthreads();

  const int lane  = tid & 31;
  const int wave  = tid >> 5;
  const int m     = lane & 15;        // A row within 16-row tile
  const int khalf = (lane >> 4) * 2;  // 0 for lanes 0-15, 2 for lanes 16-31
  const long rowBase = (long)blockIdx.x * 128 + wave * 16;
  const float* arow = input + (rowBase + m) * (long)kDimIn + khalf;
  const v2f* bLane = bTab + lane;

  v8f acc = {};
#pragma unroll 8
  for (int c = 0; c < kChunks; ++c) {
    v2f av;
    av.x = __builtin_nontemporal_load(arow + 4 * c);
    av.y = __builtin_nontemporal_load(arow + 4 * c + 1);
    const v2f bv = bLane[c * 32];
    acc = __builtin_amdgcn_wmma_f32_16x16x4_f32(false, av, false, bv,
                                                (short)0, acc, false, false);
  }

  // D layout: VGPR j -> lanes 0-15: (M=j, N=lane); lanes 16-31: (M=j+8).
  // Only columns N=0 (Cin) and N=1 (Sin) are meaningful.
  if (m < 2) {
    const long r0 = rowBase + (lane >> 4) * 8;
    float* dst = (m == 0) ? Cin : Sin;
#pragma unroll
    for (int j = 0; j < 8; ++j) dst[r0 + j] = acc[j];
  }
}

// ---------------------------------------------------------------------------
// Phase 2: recurrence. W_h_ah is rank-2 too, so the per-step matmul is two
// dot products per batch row. One block (8 waves) per batch row; 2 neurons
// per thread. Output GEMM (W_h_y, 2x512) is fused into the same per-step
// reduction. Only cross-wave dependency is a 4-float reduction -> shfl_xor
// within wave32 + double-buffered LDS (one barrier per step).
// ---------------------------------------------------------------------------
__global__ __launch_bounds__(256) void recurrence_kernel(
    const float* __restrict__ noise, const float* __restrict__ topp,
    const float* __restrict__ Why, const float* __restrict__ Cin,
    const float* __restrict__ Sin, float* __restrict__ out,
    float* __restrict__ hstore) {
  const int b    = blockIdx.x;
  const int tid  = threadIdx.x;
  const int lane = tid & 31;
  const int wave = tid >> 5;
  const int r0   = tid * 2;

  const float tp0 = topp[0];
  const float tp1 = topp[1];

  const float th0 = (float)r0 * kTwoPiOverNet;
  const float th1 = (float)(r0 + 1) * kTwoPiOverNet;
  const float c0 = cosf(th0), s0 = sinf(th0);
  const float c1 = cosf(th1), s1 = sinf(th1);
  const float w00 = Why[r0],        w01 = Why[r0 + 1];
  const float w10 = Why[kNet + r0], w11 = Why[kNet + r0 + 1];

  const float* nz   = noise + (long)b * kT * kNet;
  float*       hs   = hstore + (long)b * kT * kNet;
  const float* cinb = Cin + b * kT;
  const float* sinb = Sin + b * kT;
  float*       outb = out + (long)b * kT * kDimOut;

  __shared__ v4f red[2][8];

  float ahx = 0.0f, ahy = 0.0f;   // ah state for this thread's two neurons
  float Sc = 0.0f, Ss = 0.0f;     // sum(h*cos), sum(h*sin) from previous h

  for (int t = 0; t < kT; ++t) {
    // ah += a * (-ah + tp0*(cos*(Sc+Cin) + sin*(Ss+Sin)) + tp1)
    const float u = tp0 * (Sc + cinb[t]);
    const float v = tp0 * (Ss + sinb[t]);
    ahx = ahx + kA * (c0 * u + s0 * v + tp1 - ahx);
    ahy = ahy + kA * (c1 * u + s1 * v + tp1 - ahy);

    const v2f nzv =
        __builtin_nontemporal_load((const v2f*)(nz + (long)t * kNet + r0));
    if (t + 1 < kT)
      __builtin_prefetch(nz + (long)(t + 1) * kNet + r0, 0, 0);

    const float hx = fmaxf(tanhf(ahx), 0.0f) + nzv.x;
    const float hy = fmaxf(tanhf(ahy), 0.0f) + nzv.y;

    v2f hv; hv.x = hx; hv.y = hy;
    __builtin_nontemporal_store(hv, (v2f*)(hs + (long)t * kNet + r0));

    // per-thread partials: h.cos, h.sin, h.w_y0, h.w_y1
    v4f p;
    p.x = hx * c0 + hy * c1;
    p.y = hx * s0 + hy * s1;
    p.z = hx * w00 + hy * w01;
    p.w = hx * w10 + hy * w11;
#pragma unroll
    for (int off = 16; off >= 1; off >>= 1) {
      p.x += __shfl_xor(p.x, off, 32);
      p.y += __shfl_xor(p.y, off, 32);
      p.z += __shfl_xor(p.z, off, 32);
      p.w += __shfl_xor(p.w, off, 32);
    }
    if (lane == 0) red[t & 1][wave] = p;
    __syncthreads();
    v4f tot = red[t & 1][0];
#pragma unroll
    for (int wgi = 1; wgi < 8; ++wgi) tot += red[t & 1][wgi];
    Sc = tot.x;
    Ss = tot.y;
    if (tid == 0) {
      outb[2 * t]     = tot.z;
      outb[2 * t + 1] = tot.w;
    }
    // double-buffered red[] -> no second barrier needed
  }
}

extern "C" void kernel_launch(void* const* d_in, const int* in_sizes, int n_in,
                              void* d_out, int out_size, void* d_ws,
                              size_t ws_size, hipStream_t stream) {
  (void)in_sizes; (void)n_in; (void)out_size; (void)ws_size;
  const float* input = (const float*)d_in[0];  // [B,T,513]
  const float* noise = (const float*)d_in[1];  // [B,T,512]
  const float* topp  = (const float*)d_in[2];  // [2]
  const float* Why   = (const float*)d_in[3];  // [2,512]

  float* out    = (float*)d_out;                          // [B,T,2]
  float* hstore = out + (size_t)kB * kT * kDimOut;        // [B,T,512]

  float* Cin = (float*)d_ws;                              // [B*T]
  float* Sin = Cin + (size_t)kB * kT;                     // [B*T]

  const int rows = kB * kT;                               // 76800
  inproj_wmma_kernel<<<dim3(rows / 128), dim3(256), 0, stream>>>(input, Cin,
                                                                 Sin);
  recurrence_kernel<<<dim3(kB), dim3(256), 0, stream>>>(noise, topp, Why, Cin,
                                                        Sin, out, hstore);
}